// BarycentricCoordinates_7447473291704
// MI455X (gfx1250) — compile-verified
//
#include <hip/hip_runtime.h>
#include <math.h>

typedef __attribute__((ext_vector_type(2))) float v2f;
typedef __attribute__((ext_vector_type(8))) float v8f;

#define NVIEW 256
#define NP    16      // projection points per view
#define NT    560     // C(16,3) triangles
#define NTP   40      // template points (5*8)
#define NTILE 35      // 560/16 triangle tiles
#define BLK   96      // 3 waves: one 16-row M-tile each

__global__ __launch_bounds__(BLK) void bary_kernel(
    const float* __restrict__ tmpl,    // 40*2
    const float* __restrict__ proj,    // 256*16*2
    float* __restrict__ outBc,         // 256*40*3 (f32)
    int*   __restrict__ outIdx)        // 256*40*3 (i32)
{
  __shared__ float sTx[NTP], sTy[NTP];
  __shared__ float sPx[NP],  sPy[NP];
  __shared__ int   sIa[NT], sIb[NT], sIc[NT];
  __shared__ float sV0x[NT], sV0y[NT], sV1x[NT], sV1y[NT];
  __shared__ float sD00[NT], sD01[NT], sD11[NT], sDnm[NT], sS0[NT], sS1[NT];
  __shared__ int   sOk[NT];
  __shared__ float sDist[NTP * NP];

  const int v      = blockIdx.x;
  const int tid    = threadIdx.x;
  const int lane   = tid & 31;
  const int wv     = tid >> 5;     // 0..2 : M-tile of template rows
  const int lane16 = lane & 15;
  const int half   = lane >> 4;

  // ---- Phase 0: stage template + this view's projections into LDS ----
  for (int i = tid; i < NTP; i += BLK) { sTx[i] = tmpl[2*i]; sTy[i] = tmpl[2*i+1]; }
  for (int i = tid; i < NP;  i += BLK) { sPx[i] = proj[(v*NP+i)*2]; sPy[i] = proj[(v*NP+i)*2+1]; }
  __syncthreads();

  // ---- Phase 1: per-triangle params (f32) + incircle predicate (f64, contraction off) ----
  for (int t = tid; t < NT; t += BLK) {
    // unrank t -> lexicographic (a<b<c) combination of 0..15
    int r = t, a = 0;
    for (; a < NP; ++a) { int cnt = (15-a)*(14-a)/2; if (r < cnt) break; r -= cnt; }
    int b = a + 1;
    for (; b < NP; ++b) { int cnt = 15-b; if (r < cnt) break; r -= cnt; }
    int c = b + 1 + r;
    sIa[t] = a; sIb[t] = b; sIc[t] = c;

    float Ax = sPx[a], Ay = sPy[a], Bx = sPx[b], By = sPy[b], Cx = sPx[c], Cy = sPy[c];
    float v0x = Cx-Ax, v0y = Cy-Ay, v1x = Bx-Ax, v1y = By-Ay;
    sV0x[t]=v0x; sV0y[t]=v0y; sV1x[t]=v1x; sV1y[t]=v1y;
    float d00 = v0x*v0x + v0y*v0y;
    float d01 = v0x*v1x + v0y*v1y;
    float d11 = v1x*v1x + v1y*v1y;
    sD00[t]=d00; sD01[t]=d01; sD11[t]=d11;
    sDnm[t] = 1.0f / (d00*d11 - d01*d01);
    sS0[t]  = v0x*Ax + v0y*Ay;
    sS1[t]  = v1x*Ax + v1y*Ay;

    int cnt_in = 0;
    {
      #pragma clang fp contract(off)
      double dAx=Ax, dAy=Ay, dBx=Bx, dBy=By, dCx=Cx, dCy=Cy;
      for (int q = 0; q < NP; ++q) {
        double px = (double)sPx[q], py = (double)sPy[q];
        double ax = dAx-px, ay = dAy-py;
        double bx = dBx-px, by = dBy-py;
        double cx = dCx-px, cy = dCy-py;
        double az = ax*ax + ay*ay, bz = bx*bx + by*by, cz = cx*cx + cy*cy;
        double det = ax*by*cz + ay*bz*cx + az*bx*cy
                   - az*by*cx - ay*bx*cz - ax*bz*cy;
        cnt_in += (det > 0.0) ? 1 : 0;
      }
    }
    sOk[t] = (cnt_in <= 3) ? 1 : 0;
  }

  // ---- Phase 2: template<->point distance table (reused by all 560 triangles) ----
  for (int i = tid; i < NTP*NP; i += BLK) {
    int p = i >> 4, q = i & 15;
    float dx = sTx[p]-sPx[q], dy = sTy[p]-sPy[q];
    sDist[i] = sqrtf(dx*dx + dy*dy);
  }
  __syncthreads();

  // ---- Phase 3: WMMA f32 16x16x4 GEMMs M0=T·v0, M1=T·v1, fused argmin ----
  // A (16x4): VGPR0 = K0 rows (lanes 0-15) / K2 (zeros), VGPR1 = K1 / K3 (zeros)
  const int pA = wv*16 + lane16;
  v2f aOp;
  aOp[0] = (half == 0 && pA < NTP) ? sTx[pA] : 0.0f;   // K=0 : x
  aOp[1] = (half == 0 && pA < NTP) ? sTy[pA] : 0.0f;   // K=1 : y

  float bestD[8]; int bestT[8];
  #pragma unroll
  for (int j = 0; j < 8; ++j) { bestD[j] = __builtin_inff(); bestT[j] = 0; }

  for (int tile = 0; tile < NTILE; ++tile) {
    const int t = tile*16 + lane16;
    v2f b0, b1;                         // B (4x16): lanes = N (triangle), VGPRs = K
    b0[0] = (half == 0) ? sV0x[t] : 0.0f;
    b0[1] = (half == 0) ? sV0y[t] : 0.0f;
    b1[0] = (half == 0) ? sV1x[t] : 0.0f;
    b1[1] = (half == 0) ? sV1y[t] : 0.0f;
    v8f cz = {};
    v8f D0 = __builtin_amdgcn_wmma_f32_16x16x4_f32(false, aOp, false, b0, (short)0, cz, false, false);
    v8f D1 = __builtin_amdgcn_wmma_f32_16x16x4_f32(false, aOp, false, b1, (short)0, cz, false, false);

    const float d00 = sD00[t], d01 = sD01[t], d11 = sD11[t], dnm = sDnm[t];
    const float s0 = sS0[t], s1 = sS1[t];
    const int ia = sIa[t], ib = sIb[t], ic = sIc[t];
    const int okT = sOk[t];

    #pragma unroll
    for (int j = 0; j < 8; ++j) {
      const int p  = wv*16 + j + half*8;           // template row of this D element
      const int pc = (p < NTP) ? p : 0;
      float d02 = D0[j] - s0;
      float d12 = D1[j] - s1;
      float w2 = (d11*d02 - d01*d12) * dnm;
      float w1 = (d00*d12 - d01*d02) * dnm;
      float w0 = 1.0f - w2 - w1;
      // NaN must NOT flag bad here (matches reference semantics: NaN>1 / NaN<0 are false)
      bool bcBad = (w0 > 1.0f) | (w0 < 0.0f) | (w1 > 1.0f) | (w1 < 0.0f) | (w2 > 1.0f) | (w2 < 0.0f);
      bool masked = (okT == 0) | bcBad | (p >= NTP);
      float dist = sDist[pc*16+ia] + sDist[pc*16+ib] + sDist[pc*16+ic];
      float dd = masked ? __builtin_inff() : dist;
      if (dd < bestD[j]) { bestD[j] = dd; bestT[j] = t; }  // strict: keeps smallest t on ties
    }
  }

  // ---- Phase 4: width-16 cross-lane argmin (tie-break: smaller triangle index) ----
  #pragma unroll
  for (int j = 0; j < 8; ++j) {
    float bd = bestD[j]; int bt = bestT[j];
    #pragma unroll
    for (int off = 8; off >= 1; off >>= 1) {
      float od = __shfl_xor(bd, off, 16);
      int   ot = __shfl_xor(bt, off, 16);
      if (od < bd || (od == bd && ot < bt)) { bd = od; bt = ot; }
    }
    bestD[j] = bd; bestT[j] = bt;
  }

  // ---- Phase 5: leaders (lane 0 of each half) recompute winners & store ----
  if (lane16 == 0) {
    #pragma unroll
    for (int j = 0; j < 8; ++j) {
      const int p = wv*16 + j + half*8;
      if (p >= NTP) continue;
      float w0 = 0.0f, w1 = 0.0f, w2 = 0.0f;
      int ia = 0, ib = 0, ic = 0;
      if (bestD[j] < __builtin_inff()) {           // not all-masked
        const int t = bestT[j];
        ia = sIa[t]; ib = sIb[t]; ic = sIc[t];
        float v2x = sTx[p] - sPx[ia];
        float v2y = sTy[p] - sPy[ia];
        float d02 = sV0x[t]*v2x + sV0y[t]*v2y;
        float d12 = sV1x[t]*v2x + sV1y[t]*v2y;
        w2 = (sD11[t]*d02 - sD01[t]*d12) * sDnm[t];
        w1 = (sD00[t]*d12 - sD01[t]*d02) * sDnm[t];
        w0 = 1.0f - w2 - w1;
        if (!(isfinite(w0) && isfinite(w1) && isfinite(w2))) {  // NaN/inf fallback
          w0 = w1 = w2 = 0.0f; ia = ib = ic = 0;
        }
      }
      const long base = ((long)v*NTP + p) * 3;
      outBc[base+0] = w0; outBc[base+1] = w1; outBc[base+2] = w2;
      outIdx[base+0] = ia; outIdx[base+1] = ib; outIdx[base+2] = ic;
    }
  }
}

extern "C" void kernel_launch(void* const* d_in, const int* in_sizes, int n_in,
                              void* d_out, int out_size, void* d_ws, size_t ws_size,
                              hipStream_t stream) {
  const float* tmpl = (const float*)d_in[0];   // (5,8,2) f32
  const float* proj = (const float*)d_in[1];   // (256,16,2) f32
  float* outBc = (float*)d_out;                                  // (256,5,8,3) f32
  int*   outIdx = (int*)((float*)d_out + (size_t)NVIEW*NTP*3);   // (256,5,8,3) i32
  bary_kernel<<<dim3(NVIEW), dim3(BLK), 0, stream>>>(tmpl, proj, outBc, outIdx);
}